// NoiseLayer_69166153335174
// MI455X (gfx1250) — compile-verified
//
#include <hip/hip_runtime.h>
#include <hip/hip_bf16.h>
#include <math.h>

// ---------------------------------------------------------------------------
// out[b] = U @ X[b] @ U^H
// Kernel 1: build U (per-column independent scan, no barriers) and emit it
//           pre-swizzled into WMMA A/B fragment order (F planes).
// Kernel 2: per-batch fused complex GEMM pair with V_WMMA_F32_16X16X4_F32.
//           K runs exactly 100 (no dead zero-B WMMAs). X is staged transposed
//           in LDS so every B-frag of GEMM1 is one aligned ds_read_b64; every
//           U fragment is one coalesced global_load_b64. Two output tiles per
//           wave -> 4 independent accumulator chains per k-step.
// ---------------------------------------------------------------------------

#define DIM    100
#define PAD    112           // 7 * 16 (M/N padding only; K is exactly 100)
#define NT     7             // 16x16 tiles per dimension
#define KSTEPS 25            // K = 100 = 25 * 4
#define LDX    102           // Xt LDS stride (even: 8B-aligned b64; bank-safe)
#define LDT    114           // T  LDS stride (even: 8B-aligned b64; bank-safe)
#define FPANEL (KSTEPS * 64) // floats per tile-row panel in F (25 ks * 32 lanes * 2)

typedef __attribute__((ext_vector_type(2))) float v2f;
typedef __attribute__((ext_vector_type(8))) float v8f;

// fp32 WMMA: D = A(16x4) * B(4x16) + C   (F32 has no A/B NEG modifier)
__device__ __forceinline__ v8f wmma_f32(v2f a, v2f b, v8f c) {
  return __builtin_amdgcn_wmma_f32_16x16x4_f32(false, a, false, b, (short)0, c,
                                               false, false);
}

// ---------------------------------------------------------------------------
// Kernel 1: build U, then write fragment-order planes Fr/Fi into ws:
//   F[ti][ks][lane] = { U[m][ka], U[m][ka+1] },  m  = ti*16 + (lane&15)
//                                                ka = ks*4 + 2*(lane>>4)
// (zero-padded for m >= 100). Because U^T[k][n] = U[n][k], the same planes
// serve as GEMM1 A-frags (index by ti) and GEMM2 B-frags (index by tj).
// ---------------------------------------------------------------------------
__global__ __launch_bounds__(128) void build_unitary_kernel(
    const float* __restrict__ theta, const float* __restrict__ phi,
    const float* __restrict__ diag, float* __restrict__ ws)
{
  __shared__ float Ur[DIM * DIM];
  __shared__ float Ui[DIM * DIM];
  const int c = threadIdx.x;

  if (c < DIM) {
    for (int r = 0; r < DIM; ++r) {            // U0 = I
      Ur[r * DIM + c] = (r == c) ? 1.0f : 0.0f;
      Ui[r * DIM + c] = 0.0f;
    }
    int k = 0;                                  // Clements mesh scan
    for (int layer = 0; layer < DIM; ++layer) {
      for (int m = (layer & 1); m < DIM - 1; m += 2, ++k) {
        const int n = m + 1;
        float sth, cth, sph, cph;
        __sincosf(theta[k], &sth, &cth);
        __sincosf(phi[k],   &sph, &cph);
        const float rmr = Ur[m * DIM + c], rmi = Ui[m * DIM + c];
        const float rnr = Ur[n * DIM + c], rni = Ui[n * DIM + c];
        const float er = cph * rmr - sph * rmi;   // e^{i ph} * rm
        const float ei = cph * rmi + sph * rmr;
        Ur[m * DIM + c] = cth * er - sth * rnr;
        Ui[m * DIM + c] = cth * ei - sth * rni;
        Ur[n * DIM + c] = sth * er + cth * rnr;
        Ui[n * DIM + c] = sth * ei + cth * rni;
      }
    }
    for (int r = 0; r < DIM; ++r) {             // row phases e^{i diag[r]}
      float sd, cd;
      __sincosf(diag[r], &sd, &cd);
      const float ur = Ur[r * DIM + c], ui = Ui[r * DIM + c];
      Ur[r * DIM + c] = cd * ur - sd * ui;
      Ui[r * DIM + c] = cd * ui + sd * ur;
    }
  }
  __syncthreads();

  float* Frp = ws;                          // NT*KSTEPS*32 float2 = 11200 floats
  float* Fip = ws + NT * FPANEL;
  for (int idx = threadIdx.x; idx < NT * KSTEPS * 32; idx += blockDim.x) {
    const int lane = idx & 31;
    const int ks   = (idx >> 5) % KSTEPS;
    const int ti   = idx / (KSTEPS * 32);
    const int m    = ti * 16 + (lane & 15);
    const int ka   = ks * 4 + 2 * (lane >> 4);      // ka+1 <= 99 always
    float r0 = 0.f, r1 = 0.f, i0 = 0.f, i1 = 0.f;
    if (m < DIM) {
      r0 = Ur[m * DIM + ka];  r1 = Ur[m * DIM + ka + 1];
      i0 = Ui[m * DIM + ka];  i1 = Ui[m * DIM + ka + 1];
    }
    Frp[2 * idx] = r0;  Frp[2 * idx + 1] = r1;
    Fip[2 * idx] = i0;  Fip[2 * idx + 1] = i1;
  }
}

// ---------------------------------------------------------------------------
// Kernel 2: one workgroup (8 waves) per batch element.
// ---------------------------------------------------------------------------
__global__ __launch_bounds__(256) void uxuh_kernel(
    const float* __restrict__ xre, const float* __restrict__ xim,
    const float* __restrict__ ws, float* __restrict__ out, int nbatch)
{
  __shared__ float Xtr[PAD * LDX];   // X transposed: Xtr[col][row]
  __shared__ float Xti[PAD * LDX];
  __shared__ float Tr [PAD * LDT];   // T = U * X
  __shared__ float Ti [PAD * LDT];

  const float* __restrict__ Frp = ws;
  const float* __restrict__ Fip = ws + NT * FPANEL;

  const int b = blockIdx.x;
  if (b >= nbatch) return;
  const float* __restrict__ Xr = xre + (size_t)b * DIM * DIM;
  const float* __restrict__ Xi = xim + (size_t)b * DIM * DIM;

  const int tid   = threadIdx.x;
  const int lane  = tid & 31;
  const int wave  = tid >> 5;          // uniform per wave
  const int mloc  = lane & 15;
  const int khalf = lane >> 4;
  const int foff0 = lane * 2;          // per-lane float offset inside a ks group

  // ---- stage X transposed into LDS (coalesced global, conflict-free LDS) ----
  for (int idx = tid; idx < DIM * DIM; idx += 256) {
    const int r = idx / DIM, c = idx - r * DIM;
    Xtr[c * LDX + r] = Xr[idx];
    Xti[c * LDX + r] = Xi[idx];
  }
  // zero-pad rows (= cols of X) 100..111 over the read range ka in [0,100)
  for (int idx = tid; idx < (PAD - DIM) * DIM; idx += 256) {
    const int rr = DIM + idx / DIM, cc = idx - (idx / DIM) * DIM;
    Xtr[rr * LDX + cc] = 0.f;
    Xti[rr * LDX + cc] = 0.f;
  }
  __syncthreads();

  // ================= GEMM1: T = U * X =================
  // Pair slots: ti in 0..6, p in 0..2 -> (tj0, tj1) = (2p, 2p+1)
  for (int slot = wave; slot < NT * 3; slot += 8) {
    const int ti = slot / 3, p = slot - 3 * (slot / 3);
    const int col0 = (2 * p) * 16 + mloc;
    const int col1 = col0 + 16;
    const float* fR = Frp + ti * FPANEL;
    const float* fI = Fip + ti * FPANEL;
    v8f ar0 = {}, ai0 = {}, ar1 = {}, ai1 = {};
    for (int ks = 0; ks < KSTEPS; ++ks) {
      const int fo = ks * 64 + foff0;
      const int ka = ks * 4 + 2 * khalf;
      const v2f Ar  = *(const v2f*)(fR + fo);
      const v2f Ai  = *(const v2f*)(fI + fo);
      const v2f nAi = { -Ai.x, -Ai.y };
      const v2f Br0 = *(const v2f*)(&Xtr[col0 * LDX + ka]);
      const v2f Bi0 = *(const v2f*)(&Xti[col0 * LDX + ka]);
      const v2f Br1 = *(const v2f*)(&Xtr[col1 * LDX + ka]);
      const v2f Bi1 = *(const v2f*)(&Xti[col1 * LDX + ka]);
      ar0 = wmma_f32(Ar,  Br0, ar0);
      ai0 = wmma_f32(Ar,  Bi0, ai0);
      ar1 = wmma_f32(Ar,  Br1, ar1);
      ai1 = wmma_f32(Ar,  Bi1, ai1);
      ar0 = wmma_f32(nAi, Bi0, ar0);
      ai0 = wmma_f32(Ai,  Br0, ai0);
      ar1 = wmma_f32(nAi, Bi1, ar1);
      ai1 = wmma_f32(Ai,  Br1, ai1);
    }
    const int row0 = ti * 16 + 8 * khalf;
#pragma unroll
    for (int r = 0; r < 8; ++r) {
      const int rr = row0 + r;
      Tr[rr * LDT + col0] = ar0[r];  Ti[rr * LDT + col0] = ai0[r];
      Tr[rr * LDT + col1] = ar1[r];  Ti[rr * LDT + col1] = ai1[r];
    }
  }
  // Single slots: tj = 6
  for (int ti = wave; ti < NT; ti += 8) {
    const int col0 = 6 * 16 + mloc;
    const float* fR = Frp + ti * FPANEL;
    const float* fI = Fip + ti * FPANEL;
    v8f ar0 = {}, ai0 = {};
    for (int ks = 0; ks < KSTEPS; ++ks) {
      const int fo = ks * 64 + foff0;
      const int ka = ks * 4 + 2 * khalf;
      const v2f Ar  = *(const v2f*)(fR + fo);
      const v2f Ai  = *(const v2f*)(fI + fo);
      const v2f nAi = { -Ai.x, -Ai.y };
      const v2f Br0 = *(const v2f*)(&Xtr[col0 * LDX + ka]);
      const v2f Bi0 = *(const v2f*)(&Xti[col0 * LDX + ka]);
      ar0 = wmma_f32(Ar,  Br0, ar0);
      ai0 = wmma_f32(Ar,  Bi0, ai0);
      ar0 = wmma_f32(nAi, Bi0, ar0);
      ai0 = wmma_f32(Ai,  Br0, ai0);
    }
    const int row0 = ti * 16 + 8 * khalf;
#pragma unroll
    for (int r = 0; r < 8; ++r) {
      const int rr = row0 + r;
      Tr[rr * LDT + col0] = ar0[r];  Ti[rr * LDT + col0] = ai0[r];
    }
  }
  __syncthreads();

  // ================= GEMM2: C = T * U^H =================
  // U^H[k][n] = conj(U[n][k]) -> Br = F_r[tj], Bi carries the minus sign:
  //   Cr = Tr*Utr + Ti*Uti ;  Ci = Ti*Utr - Tr*Uti
  for (int slot = wave; slot < NT * 3; slot += 8) {
    const int ti = slot / 3, p = slot - 3 * (slot / 3);
    const int tj0 = 2 * p, tj1 = tj0 + 1;
    const int row = ti * 16 + mloc;
    const float* fR0 = Frp + tj0 * FPANEL;
    const float* fI0 = Fip + tj0 * FPANEL;
    const float* fR1 = Frp + tj1 * FPANEL;
    const float* fI1 = Fip + tj1 * FPANEL;
    v8f cr0 = {}, ci0 = {}, cr1 = {}, ci1 = {};
    for (int ks = 0; ks < KSTEPS; ++ks) {
      const int fo = ks * 64 + foff0;
      const int ka = ks * 4 + 2 * khalf;
      const v2f Atr  = *(const v2f*)(&Tr[row * LDT + ka]);
      const v2f Ati  = *(const v2f*)(&Ti[row * LDT + ka]);
      const v2f nAtr = { -Atr.x, -Atr.y };
      const v2f Br0 = *(const v2f*)(fR0 + fo);
      const v2f Bi0 = *(const v2f*)(fI0 + fo);
      const v2f Br1 = *(const v2f*)(fR1 + fo);
      const v2f Bi1 = *(const v2f*)(fI1 + fo);
      cr0 = wmma_f32(Atr,  Br0, cr0);
      ci0 = wmma_f32(Ati,  Br0, ci0);
      cr1 = wmma_f32(Atr,  Br1, cr1);
      ci1 = wmma_f32(Ati,  Br1, ci1);
      cr0 = wmma_f32(Ati,  Bi0, cr0);
      ci0 = wmma_f32(nAtr, Bi0, ci0);
      cr1 = wmma_f32(Ati,  Bi1, cr1);
      ci1 = wmma_f32(nAtr, Bi1, ci1);
    }
    const int row0 = ti * 16 + 8 * khalf;
    const int c0 = tj0 * 16 + mloc;     // <= 95 for pairs: always < DIM
    const int c1 = c0 + 16;
#pragma unroll
    for (int r = 0; r < 8; ++r) {
      const int rr = row0 + r;
      if (rr < DIM) {
        float2 v0; v0.x = cr0[r]; v0.y = ci0[r];
        float2 v1; v1.x = cr1[r]; v1.y = ci1[r];
        *(float2*)(out + (((size_t)b * DIM + rr) * DIM + c0) * 2) = v0;
        *(float2*)(out + (((size_t)b * DIM + rr) * DIM + c1) * 2) = v1;
      }
    }
  }
  // Singles: tj = 6 (cols 96..111, guarded)
  for (int ti = wave; ti < NT; ti += 8) {
    const int row = ti * 16 + mloc;
    const float* fR0 = Frp + 6 * FPANEL;
    const float* fI0 = Fip + 6 * FPANEL;
    v8f cr0 = {}, ci0 = {};
    for (int ks = 0; ks < KSTEPS; ++ks) {
      const int fo = ks * 64 + foff0;
      const int ka = ks * 4 + 2 * khalf;
      const v2f Atr  = *(const v2f*)(&Tr[row * LDT + ka]);
      const v2f Ati  = *(const v2f*)(&Ti[row * LDT + ka]);
      const v2f nAtr = { -Atr.x, -Atr.y };
      const v2f Br0 = *(const v2f*)(fR0 + fo);
      const v2f Bi0 = *(const v2f*)(fI0 + fo);
      cr0 = wmma_f32(Atr,  Br0, cr0);
      ci0 = wmma_f32(Ati,  Br0, ci0);
      cr0 = wmma_f32(Ati,  Bi0, cr0);
      ci0 = wmma_f32(nAtr, Bi0, ci0);
    }
    const int row0 = ti * 16 + 8 * khalf;
    const int c0 = 6 * 16 + mloc;
#pragma unroll
    for (int r = 0; r < 8; ++r) {
      const int rr = row0 + r;
      if (rr < DIM && c0 < DIM) {
        float2 v0; v0.x = cr0[r]; v0.y = ci0[r];
        *(float2*)(out + (((size_t)b * DIM + rr) * DIM + c0) * 2) = v0;
      }
    }
  }
}

extern "C" void kernel_launch(void* const* d_in, const int* in_sizes, int n_in,
                              void* d_out, int out_size, void* d_ws, size_t ws_size,
                              hipStream_t stream) {
  const float* x_re  = (const float*)d_in[0];
  const float* x_im  = (const float*)d_in[1];
  const float* theta = (const float*)d_in[2];
  const float* phi   = (const float*)d_in[3];
  const float* diag  = (const float*)d_in[4];
  float* out = (float*)d_out;            // complex64 -> interleaved (re, im)
  float* ws  = (float*)d_ws;             // needs 2*NT*KSTEPS*64*4 B ~= 90 KB

  const int nbatch = in_sizes[0] / (DIM * DIM);   // 2048

  build_unitary_kernel<<<1, 128, 0, stream>>>(theta, phi, diag, ws);
  uxuh_kernel<<<nbatch, 256, 0, stream>>>(x_re, x_im, ws, out, nbatch);
}